// PerLayerTopK_70239895159490
// MI455X (gfx1250) — compile-verified
//
#include <hip/hip_runtime.h>
#include <stdint.h>

// Per-row top-k (k=64) over rows of D=16384 fp32.
// One 1024-thread workgroup per row; row staged into LDS once via CDNA5
// async global->LDS loads; 4-pass MSB radix select in LDS; single writeback.

#define DIMD 16384
#define NT   1024
#define NHIST 8

__global__ __launch_bounds__(NT)
void PerLayerTopK_kernel(const float* __restrict__ in,
                         const int* __restrict__ kp,
                         float* __restrict__ out)
{
    __shared__ uint32_t keys[DIMD];        // 64 KB: order-transformed keys of one row
    __shared__ uint32_t hist[NHIST][256];  // 8 KB : privatized histograms
    __shared__ uint32_t s_prefix;
    __shared__ uint32_t s_remaining;
    __shared__ uint32_t s_eqcnt;
    __shared__ int      s_cut;

    const int tid = threadIdx.x;
    const int row = blockIdx.x;
    const int K   = *kp;

    const float* src = in  + (size_t)row * DIMD;
    float*       dst = out + (size_t)row * DIMD;

    // ---- Stage the row into LDS with async global->LDS B128 copies ----
    {
        uint32_t lds0 = (uint32_t)(uintptr_t)&keys[0];  // LDS byte offset
#pragma unroll
        for (int j = 0; j < DIMD / (NT * 4); ++j) {
            uint32_t goff = (uint32_t)(tid + j * NT) * 16u;  // byte offset in row
            uint32_t loff = lds0 + goff;                     // LDS dest offset
            asm volatile("global_load_async_to_lds_b128 %0, %1, %2"
                         :
                         : "v"(loff), "v"(goff), "s"(src)
                         : "memory");
        }
        asm volatile("s_wait_asynccnt 0" ::: "memory");
    }
    __syncthreads();

    // ---- In-place monotonic transform: float bits -> totally ordered uint ----
    for (int i = tid; i < DIMD; i += NT) {
        uint32_t u = keys[i];
        keys[i] = u ^ (uint32_t)(((int32_t)u >> 31) | 0x80000000);
    }
    if (tid == 0) { s_prefix = 0u; s_remaining = (uint32_t)K; }
    __syncthreads();

    // ---- 4-pass MSB-first radix select: find the K-th largest key ----
    uint32_t prefix = 0u;
#pragma unroll
    for (int pass = 0; pass < 4; ++pass) {
        const int shift = 24 - 8 * pass;
        for (int b = tid; b < NHIST * 256; b += NT) ((uint32_t*)hist)[b] = 0u;
        __syncthreads();

        const uint32_t mask = (pass == 0) ? 0u : (0xFFFFFFFFu << (shift + 8));
        uint32_t* hp = hist[(tid >> 7) & (NHIST - 1)];
        for (int i = tid; i < DIMD; i += NT) {
            uint32_t kk = keys[i];
            if ((kk & mask) == prefix)
                atomicAdd(&hp[(kk >> shift) & 0xFFu], 1u);
        }
        __syncthreads();

        if (tid < 256) {           // reduce privatized histograms
            uint32_t sum = 0;
#pragma unroll
            for (int h = 0; h < NHIST; ++h) sum += hist[h][tid];
            hist[0][tid] = sum;
        }
        __syncthreads();

        if (tid == 0) {            // pick the bin holding the K-th largest
            uint32_t rem = s_remaining, cum = 0;
            int b = 255;
            for (; b > 0; --b) {
                uint32_t c = hist[0][b];
                if (cum + c >= rem) break;
                cum += c;
            }
            s_prefix |= ((uint32_t)b) << shift;
            s_remaining = rem - cum;
            if (pass == 3) s_eqcnt = hist[0][b];
        }
        __syncthreads();
        prefix = s_prefix;
    }

    const uint32_t T  = s_prefix;     // exact K-th largest key
    const uint32_t nt = s_remaining;  // how many keys == T belong to the top-K

    // ---- Tie cut index (earliest-index ties win); slow path is rare ----
    if (tid == 0) {
        int cut = DIMD;
        if (s_eqcnt > nt) {
            uint32_t c = 0;
            for (int i = 0; i < DIMD; ++i)
                if (keys[i] == T) { if (++c > nt) { cut = i; break; } }
        }
        s_cut = cut;
    }
    __syncthreads();
    const int cut = s_cut;

    // ---- Writeback: keep top-K values, zero everything else ----
    const uint4* kv = (const uint4*)keys;
    float4*      dv = (float4*)dst;
    for (int q = tid; q < DIMD / 4; q += NT) {
        uint4 kk = kv[q];
        const int base = q * 4;
        float4 o; uint32_t u;
        u = kk.x ^ ((~(uint32_t)((int32_t)kk.x >> 31)) | 0x80000000u);
        o.x = (kk.x > T || (kk.x == T && base + 0 < cut)) ? __uint_as_float(u) : 0.0f;
        u = kk.y ^ ((~(uint32_t)((int32_t)kk.y >> 31)) | 0x80000000u);
        o.y = (kk.y > T || (kk.y == T && base + 1 < cut)) ? __uint_as_float(u) : 0.0f;
        u = kk.z ^ ((~(uint32_t)((int32_t)kk.z >> 31)) | 0x80000000u);
        o.z = (kk.z > T || (kk.z == T && base + 2 < cut)) ? __uint_as_float(u) : 0.0f;
        u = kk.w ^ ((~(uint32_t)((int32_t)kk.w >> 31)) | 0x80000000u);
        o.w = (kk.w > T || (kk.w == T && base + 3 < cut)) ? __uint_as_float(u) : 0.0f;
        dv[q] = o;
    }
}

extern "C" void kernel_launch(void* const* d_in, const int* in_sizes, int n_in,
                              void* d_out, int out_size, void* d_ws, size_t ws_size,
                              hipStream_t stream) {
    (void)n_in; (void)out_size; (void)d_ws; (void)ws_size;
    const float* feats = (const float*)d_in[0];
    const int*   kptr  = (const int*)d_in[1];
    float*       out   = (float*)d_out;
    const int rows = in_sizes[0] / DIMD;  // B*L = 12288
    PerLayerTopK_kernel<<<rows, NT, 0, stream>>>(feats, kptr, out);
}